// Regressor_83923660964337
// MI455X (gfx1250) — compile-verified
//
#include <hip/hip_runtime.h>
#include <hip/hip_bf16.h>

typedef __attribute__((ext_vector_type(2))) float v2f;
typedef __attribute__((ext_vector_type(8))) float v8f;

#define N_NODES 100000
#define N_EDGES 1600000
#define HID     128
#define N_PRED  8

// ---------------------------------------------------------------- utilities
__global__ void fill_zero_kernel(float* __restrict__ p, long n) {
    long i = (long)blockIdx.x * blockDim.x + threadIdx.x;
    long stride = (long)gridDim.x * blockDim.x;
    for (; i < n; i += stride) p[i] = 0.0f;
}

// ---------------------------------------------------------------- degrees
__global__ void degree_kernel(const int* __restrict__ src, const int* __restrict__ dst,
                              float* __restrict__ deg_out, float* __restrict__ deg_in) {
    long e = (long)blockIdx.x * blockDim.x + threadIdx.x;
    if (e >= N_EDGES) return;
    atomicAdd(&deg_out[src[e]], 1.0f);
    atomicAdd(&deg_in[dst[e]], 1.0f);
}

// norm = deg > 0 ? rsqrt(deg) : 0   (in place over both deg arrays)
__global__ void norm_kernel(float* __restrict__ deg, long n) {
    long i = (long)blockIdx.x * blockDim.x + threadIdx.x;
    if (i >= n) return;
    float d = deg[i];
    deg[i] = (d > 0.0f) ? __frsqrt_rn(d) : 0.0f;
}

// ---------------------------------------------------------------- h * norm_out
__global__ void scale_rows_kernel(const float* __restrict__ h, const float* __restrict__ norm,
                                  float* __restrict__ out) {
    // one float4 per thread, 32 float4 = 1 row
    long i = (long)blockIdx.x * blockDim.x + threadIdx.x;   // float4 index
    long total = (long)N_NODES * (HID / 4);
    if (i >= total) return;
    long node = i / (HID / 4);
    float s = norm[node];
    float4 v = ((const float4*)h)[i];
    v.x *= s; v.y *= s; v.z *= s; v.w *= s;
    ((float4*)out)[i] = v;
}

// ---------------------------------------------------------------- edge scatter
// one lane handles 4 consecutive dims of one edge: wave = 1 edge (32 lanes x 4 = 128)
__global__ void spmm_scatter_kernel(const float* __restrict__ hs,
                                    const int* __restrict__ src, const int* __restrict__ dst,
                                    float* __restrict__ m) {
    long gid = (long)blockIdx.x * blockDim.x + threadIdx.x;
    long e = gid >> 5;
    if (e >= N_EDGES) return;
    int lane = (int)(gid & 31);
    int s = src[e];
    int d = dst[e];
    int dim = lane * 4;
    float4 v = *(const float4*)(hs + (long)s * HID + dim);
    float* mp = m + (long)d * HID + dim;
    atomicAdd(mp + 0, v.x);
    atomicAdd(mp + 1, v.y);
    atomicAdd(mp + 2, v.z);
    atomicAdd(mp + 3, v.w);
}

// ---------------------------------------------------------------- WMMA GEMM
// out[16 rows x 128 cols per block] = relu((m * norm_in) @ W + b)
// block = 256 threads = 8 waves; wave w computes cols [16w, 16w+16)
__global__ __launch_bounds__(256) void gemm_relu_wmma_kernel(
    const float* __restrict__ m,        // [N_NODES, HID] aggregated messages
    const float* __restrict__ norm_in,  // [N_NODES]
    const float* __restrict__ W,        // [HID, HID] row-major (K, N)
    const float* __restrict__ bias,     // [HID]
    float* __restrict__ out)            // [N_NODES, HID]
{
    __shared__ float lds[16 * 132];     // stride 132 -> conflict-free (4r+k covers 64 banks)
    const int block_row = blockIdx.x * 16;
    const int tid = threadIdx.x;

    // stage 16x128 A tile (scaled by norm_in) into LDS; 512 float4 loads / 256 threads
    for (int i = tid; i < 16 * 32; i += 256) {
        int r  = i >> 5;
        int c4 = (i & 31) * 4;
        long node = block_row + r;
        float s = norm_in[node];
        float4 v = *(const float4*)(m + node * HID + c4);
        float* dstp = &lds[r * 132 + c4];
        dstp[0] = v.x * s; dstp[1] = v.y * s; dstp[2] = v.z * s; dstp[3] = v.w * s;
    }
    __syncthreads();

    const int wave = tid >> 5;          // 0..7 -> which 16-col tile
    const int lane = tid & 31;
    const int hi   = lane >> 4;         // 0: K+0/K+1 & M rows 0-7 ; 1: K+2/K+3 & M rows 8-15
    const int nidx = lane & 15;
    const int col  = wave * 16 + nidx;

    v8f acc = {};
    #pragma unroll 4
    for (int k0 = 0; k0 < HID; k0 += 4) {
        int ka = k0 + hi * 2;
        // A fragment 16x4: lane M = lane&15; v0 = K(ka), v1 = K(ka+1)
        v2f a = *(const v2f*)(&lds[nidx * 132 + ka]);
        // B fragment 4x16: lane N = lane&15; v0 = row K(ka), v1 = row K(ka+1)
        v2f b;
        b.x = W[(ka + 0) * HID + col];
        b.y = W[(ka + 1) * HID + col];
        acc = __builtin_amdgcn_wmma_f32_16x16x4_f32(
            /*neg_a=*/false, a, /*neg_b=*/false, b,
            /*c_mod=*/(short)0, acc, /*reuse_a=*/false, /*reuse_b=*/false);
    }

    // C/D layout: VGPR r -> M = r + 8*hi, N = col
    float bb = bias[col];
    const long node0 = block_row + hi * 8;
    #pragma unroll
    for (int r = 0; r < 8; ++r) {
        float v = acc[r] + bb;
        out[(node0 + r) * HID + col] = fmaxf(v, 0.0f);
    }
}

// ---------------------------------------------------------------- mean pool partials
__global__ void colsum_kernel(const float* __restrict__ h, float* __restrict__ hg) {
    int dim = threadIdx.x;              // blockDim = 128
    float acc = 0.0f;
    for (long node = blockIdx.x; node < N_NODES; node += gridDim.x)
        acc += h[node * HID + dim];
    atomicAdd(&hg[dim], acc);
}

// ---------------------------------------------------------------- readout
__global__ void readout_kernel(const float* __restrict__ hg, const float* __restrict__ Wr,
                               const float* __restrict__ br, float* __restrict__ out) {
    int p = threadIdx.x;
    if (p >= N_PRED) return;
    const float inv_n = 1.0f / (float)N_NODES;
    float acc = 0.0f;
    for (int k = 0; k < HID; ++k)
        acc += hg[k] * inv_n * Wr[k * N_PRED + p];
    out[p] = acc + br[p];
}

// ---------------------------------------------------------------- launch
extern "C" void kernel_launch(void* const* d_in, const int* in_sizes, int n_in,
                              void* d_out, int out_size, void* d_ws, size_t ws_size,
                              hipStream_t stream) {
    const float* features = (const float*)d_in[0];
    const float* W0 = (const float*)d_in[1];
    const float* b0 = (const float*)d_in[2];
    const float* W1 = (const float*)d_in[3];
    const float* b1 = (const float*)d_in[4];
    const float* W2 = (const float*)d_in[5];
    const float* b2 = (const float*)d_in[6];
    const float* Wr = (const float*)d_in[7];
    const float* br = (const float*)d_in[8];
    const int*   src = (const int*)d_in[9];
    const int*   dst = (const int*)d_in[10];
    float* out = (float*)d_out;

    float* ws = (float*)d_ws;
    const long NH = (long)N_NODES * HID;
    float* deg_out = ws;                     // [N_NODES]  -> becomes norm_out
    float* deg_in  = ws + N_NODES;           // [N_NODES]  -> becomes norm_in
    float* bufA    = ws + 2L * N_NODES;      // [N_NODES, HID] scaled h
    float* bufB    = bufA + NH;              // [N_NODES, HID] aggregated messages
    float* bufC    = bufB + NH;              // [N_NODES, HID] layer output
    float* hg      = bufC + NH;              // [HID] pooled sum

    // zero degree arrays + pooled accumulator
    fill_zero_kernel<<<1024, 256, 0, stream>>>(deg_out, 2L * N_NODES);
    fill_zero_kernel<<<1, 128, 0, stream>>>(hg, HID);

    // degrees and symmetric norms
    degree_kernel<<<(N_EDGES + 255) / 256, 256, 0, stream>>>(src, dst, deg_out, deg_in);
    norm_kernel<<<(2 * N_NODES + 255) / 256, 256, 0, stream>>>(deg_out, 2L * N_NODES);

    const float* Ws[3] = {W0, W1, W2};
    const float* bs[3] = {b0, b1, b2};
    const float* h = features;

    const int scale_blocks = (int)(((long)N_NODES * (HID / 4) + 255) / 256);
    const int spmm_blocks  = (int)(((long)N_EDGES * 32 + 255) / 256);
    const int gemm_blocks  = N_NODES / 16;   // 100000 / 16 = 6250 exactly

    for (int l = 0; l < 3; ++l) {
        scale_rows_kernel<<<scale_blocks, 256, 0, stream>>>(h, deg_out, bufA);
        fill_zero_kernel<<<8192, 256, 0, stream>>>(bufB, NH);
        spmm_scatter_kernel<<<spmm_blocks, 256, 0, stream>>>(bufA, src, dst, bufB);
        gemm_relu_wmma_kernel<<<gemm_blocks, 256, 0, stream>>>(bufB, deg_in, Ws[l], bs[l], bufC);
        h = bufC;
    }

    colsum_kernel<<<512, 128, 0, stream>>>(bufC, hg);
    readout_kernel<<<1, 32, 0, stream>>>(hg, Wr, br, out);
}